// ContrastiveLoss_81381040325084
// MI455X (gfx1250) — compile-verified
//
#include <hip/hip_runtime.h>
#include <hip/hip_bf16.h>

typedef __attribute__((ext_vector_type(2))) float v2f;
typedef __attribute__((ext_vector_type(8))) float v8f;

#define NROWS 8192
#define DDIM  128
#define NCOLS 4096

constexpr float INV_TEMP = 1.0f / 0.3f;

#if __has_builtin(__builtin_amdgcn_wmma_f32_16x16x4_f32)
#define HAVE_WMMA_F32X4 1
#else
#define HAVE_WMMA_F32X4 0
#endif

// ---------------------------------------------------------------------------
// init: zero the two double accumulators in workspace (ws is poisoned 0xAA)
// ---------------------------------------------------------------------------
__global__ void cl_init(double* __restrict__ acc) {
  acc[0] = 0.0;  // total sum
  acc[1] = 0.0;  // matched (num) sum
}

// ---------------------------------------------------------------------------
// main: S = exp((x @ y^T)/T); reduce total and matched sums.
// One wave -> 16x64 output tile via 4x V_WMMA_F32_16X16X4_F32 per K-chunk.
// ---------------------------------------------------------------------------
__global__ void __launch_bounds__(256) cl_main(const float* __restrict__ x,
                                               const int* __restrict__ tix,
                                               const float* __restrict__ yflat,
                                               double* __restrict__ acc) {
  const int wave = (blockIdx.x << 3) + (threadIdx.x >> 5);  // 8 waves/block
  const int lane = threadIdx.x & 31;
  const int half = lane >> 4;   // 0: lanes 0-15, 1: lanes 16-31
  const int l16  = lane & 15;

  const int tile_m  = wave >> 6;   // 512 row tiles of 16
  const int tile_ng = wave & 63;   // 64 col groups of 64
  const int r0 = tile_m << 4;
  const int c0 = tile_ng << 6;

  // f32 WMMA operand layout (ISA 7.12.2):
  //   A 16x4: lane-half h holds K = {2h, 2h+1} in VGPR0/1, M = lane&15
  //   B 4x16: lane-half h holds K = {2h, 2h+1} in VGPR0/1, N = lane&15
  // Both are contiguous float2 in row-major x / y_flat.
  const float* xr = x     + (size_t)(r0 + l16) * DDIM + (half << 1);
  const float* y0 = yflat + (size_t)(c0 +  0 + l16) * DDIM + (half << 1);
  const float* y1 = y0 + 16 * DDIM;
  const float* y2 = y0 + 32 * DDIM;
  const float* y3 = y0 + 48 * DDIM;

  v8f a0 = {}, a1 = {}, a2 = {}, a3 = {};

#if HAVE_WMMA_F32X4
#pragma unroll
  for (int c = 0; c < DDIM / 4; ++c) {
    v2f av = *(const v2f*)(xr + 4 * c);
    v2f b0 = *(const v2f*)(y0 + 4 * c);
    v2f b1 = *(const v2f*)(y1 + 4 * c);
    v2f b2 = *(const v2f*)(y2 + 4 * c);
    v2f b3 = *(const v2f*)(y3 + 4 * c);
    a0 = __builtin_amdgcn_wmma_f32_16x16x4_f32(false, av, false, b0, (short)0, a0, false, false);
    a1 = __builtin_amdgcn_wmma_f32_16x16x4_f32(false, av, false, b1, (short)0, a1, false, false);
    a2 = __builtin_amdgcn_wmma_f32_16x16x4_f32(false, av, false, b2, (short)0, a2, false, false);
    a3 = __builtin_amdgcn_wmma_f32_16x16x4_f32(false, av, false, b3, (short)0, a3, false, false);
  }
#else
  // Scalar fallback (compile-probe only): lane computes its own C elements.
  // C layout emulated: acc[s][v] = dot(x[r0+v+8*half], y[c0+16*s+l16])
  {
    const float* ys[4] = {yflat + (size_t)(c0 + l16) * DDIM,
                          yflat + (size_t)(c0 + 16 + l16) * DDIM,
                          yflat + (size_t)(c0 + 32 + l16) * DDIM,
                          yflat + (size_t)(c0 + 48 + l16) * DDIM};
    for (int v = 0; v < 8; ++v) {
      const float* xv = x + (size_t)(r0 + v + (half << 3)) * DDIM;
      float d0 = 0.f, d1 = 0.f, d2 = 0.f, d3 = 0.f;
      for (int k = 0; k < DDIM; ++k) {
        float xk = xv[k];
        d0 = fmaf(xk, ys[0][k], d0);
        d1 = fmaf(xk, ys[1][k], d1);
        d2 = fmaf(xk, ys[2][k], d2);
        d3 = fmaf(xk, ys[3][k], d3);
      }
      a0[v] = d0; a1[v] = d1; a2[v] = d2; a3[v] = d3;
    }
  }
#endif

  // Row labels for this lane-half: C-matrix row = r0 + v + 8*half.
  int labs[8];
#pragma unroll
  for (int v = 0; v < 8; ++v) labs[v] = tix[r0 + v + (half << 3)];

  float tot = 0.f, num = 0.f;

#define CL_EPILOGUE(A, S)                                   \
  {                                                         \
    const int clab = (c0 + (S) * 16 + l16) & 511;           \
    _Pragma("unroll") for (int v = 0; v < 8; ++v) {         \
      float e = __expf((A)[v] * INV_TEMP);                  \
      tot += e;                                             \
      if (labs[v] == clab) num += e;                        \
    }                                                       \
  }
  CL_EPILOGUE(a0, 0)
  CL_EPILOGUE(a1, 1)
  CL_EPILOGUE(a2, 2)
  CL_EPILOGUE(a3, 3)
#undef CL_EPILOGUE

  // wave32 reduction
#pragma unroll
  for (int off = 16; off > 0; off >>= 1) {
    tot += __shfl_xor(tot, off, 32);
    num += __shfl_xor(num, off, 32);
  }
  if (lane == 0) {
    atomicAdd(&acc[0], (double)tot);
    atomicAdd(&acc[1], (double)num);
  }
}

// ---------------------------------------------------------------------------
// finalize: loss = -log(num/(num+den)) = log(total) - log(num)
// ---------------------------------------------------------------------------
__global__ void cl_final(const double* __restrict__ acc, float* __restrict__ out) {
  out[0] = (float)(log(acc[0]) - log(acc[1]));
}

extern "C" void kernel_launch(void* const* d_in, const int* in_sizes, int n_in,
                              void* d_out, int out_size, void* d_ws, size_t ws_size,
                              hipStream_t stream) {
  const float* x   = (const float*)d_in[0];   // [8192,128] f32
  const int*   tix = (const int*)d_in[1];     // [8192] i32
  const float* y   = (const float*)d_in[2];   // [512,8,128] f32 == y_flat [4096,128]
  double* acc = (double*)d_ws;
  float*  out = (float*)d_out;

  cl_init<<<1, 1, 0, stream>>>(acc);
  const int waves  = (NROWS / 16) * (NCOLS / 64);  // 32768 wave-tiles
  const int blocks = waves / 8;                    // 8 waves per 256-thread block
  cl_main<<<blocks, 256, 0, stream>>>(x, tix, y, acc);
  cl_final<<<1, 1, 0, stream>>>(acc, out);
}